// S4DKernel_80513456931080
// MI455X (gfx1250) — compile-verified
//
#include <hip/hip_runtime.h>
#include <hip/hip_bf16.h>

typedef __attribute__((ext_vector_type(2))) float v2f;
typedef __attribute__((ext_vector_type(8))) float v8f;

#define HDIM 1024
#define NH   32
#define LLEN 4096

__device__ __forceinline__ void csq(float& r, float& i) {   // (r,i) = (r,i)^2
    float t = fmaf(r, r, -(i * i));
    i = 2.0f * r * i;
    r = t;
}
__device__ __forceinline__ void cmul(float& r, float& i, float br, float bi) {
    float t = fmaf(r, br, -(i * bi));
    i = fmaf(r, bi, i * br);
    r = t;
}

// One block per head h. 4 waves / block; each wave computes 4 output tiles of
// 16(lhi) x 16(llo) via V_WMMA_F32_16X16X4_F32 with a K=64 reduction
// (32 complex modes split into {Re,Im}). All exponentials are built from
// small-argument __expf/__sincosf plus complex squaring chains -> no libm
// slow path; all control flow is wave-uniform (EXEC stays all-ones).
__global__ __launch_bounds__(128) void s4d_wmma_kernel(
    const float* __restrict__ C_ri,        // (H, 32, 2)
    const float* __restrict__ log_dt,      // (H,)
    const float* __restrict__ log_A_real,  // (H, 32)
    const float* __restrict__ A_imag,      // (H, 32)
    float* __restrict__ out)               // (H, 4096)
{
    const int h    = blockIdx.x;
    const int tid  = threadIdx.x;
    const int lane = tid & 31;
    // readfirstlane: tell the compiler `wave` is wave-uniform (scalar)
    const int wave = __builtin_amdgcn_readfirstlane(tid >> 5);
    const int idx  = lane & 15;  // A row M within tile / B column llo
    const int half = lane >> 4;  // 0 or 1
    const int off  = half * 2;   // K sub-offset inside each chunk of 4

    const float dt = __expf(log_dt[h]);

    // Per-lane state for its 16 modes: n = 4*j + off + s
    float pwr[8][2], pwi[8][2];  // running w * exp(dtA * 16*lhi), lhi = wave*64 + idx
    float sr [8][2], si [8][2];  // per-tile step e256 = exp(dtA * 256)
    float br_[8][2], bi_[8][2];  // B operand: Re(Q), -Im(Q), Q = exp(dtA * idx)

    const bool b0 = (wave & 1) != 0;
    const bool b1 = (wave & 2) != 0;

#pragma unroll
    for (int j = 0; j < 8; ++j) {
#pragma unroll
        for (int s = 0; s < 2; ++s) {
            const int n   = 4 * j + off + s;
            const float Ar = -__expf(log_A_real[h * NH + n]);
            const float Ai = A_imag[h * NH + n];
            const float zr = Ar * dt, zi = Ai * dt;          // dtA; |zi| <= ~9.8

            // em1 = exp(dtA) - 1                              (arg <= ~10 rad)
            float s1, c1;  __sincosf(zi, &s1, &c1);
            const float m1   = __expf(zr);
            const float em1r = m1 * c1 - 1.0f;
            const float em1i = m1 * s1;

            // w = 2 * C * em1 * conj(A) / |A|^2
            const float cr0 = C_ri[(h * NH + n) * 2 + 0];
            const float ci0 = C_ri[(h * NH + n) * 2 + 1];
            const float tr  = cr0 * em1r - ci0 * em1i;
            const float ti  = cr0 * em1i + ci0 * em1r;
            const float inv = 2.0f / (Ar * Ar + Ai * Ai);
            const float wre = (tr * Ar + ti * Ai) * inv;
            const float wim = (ti * Ar - tr * Ai) * inv;

            // Q = exp(dtA * idx)                              (arg <= ~147 rad)
            const float fl = (float)idx;
            float sq, cq;  __sincosf(zi * fl, &sq, &cq);
            const float mq = __expf(zr * fl);
            const float qr = mq * cq, qi = mq * sq;
            br_[j][s] =  qr;
            bi_[j][s] = -qi;

            // Q^16 = exp(dtA * 16*idx)  (4 complex squarings)
            float xr = qr, xi = qi;
            csq(xr, xi); csq(xr, xi); csq(xr, xi); csq(xr, xi);

            // e16 = exp(dtA * 16)                             (arg <= ~157 rad)
            float s16, c16;  __sincosf(zi * 16.0f, &s16, &c16);
            const float m16 = __expf(zr * 16.0f);
            float er = m16 * c16, ei = m16 * s16;
            csq(er, ei); csq(er, ei); csq(er, ei); csq(er, ei);   // e256
            sr[j][s] = er;  si[j][s] = ei;

            // pw = w * Q^16 * e1024^wave, built branchlessly from wave's bits
            float pr = wre, pi = wim;
            cmul(pr, pi, xr, xi);                                 // * Q^16
            csq(er, ei); csq(er, ei);                             // e1024
            const float m0r = b0 ? er : 1.0f, m0i = b0 ? ei : 0.0f;
            csq(er, ei);                                          // e2048
            const float m1r = b1 ? er : 1.0f, m1i = b1 ? ei : 0.0f;
            cmul(pr, pi, m0r, m0i);                               // * e1024^bit0
            cmul(pr, pi, m1r, m1i);                               // * e2048^bit1
            pwr[j][s] = pr;  pwi[j][s] = pi;
        }
    }

    // 256 lhi rows -> 16 tiles; this wave owns tiles tau = wave*4 .. wave*4+3
#pragma unroll
    for (int t = 0; t < 4; ++t) {
        const int tau = wave * 4 + t;   // scalar
        v8f c = {};

#pragma unroll
        for (int j = 0; j < 8; ++j) {
            v2f a_re, a_im, b_re, b_im;
#pragma unroll
            for (int s = 0; s < 2; ++s) {
                a_re[s] = pwr[j][s];                 // Re(w * P)
                a_im[s] = pwi[j][s];                 // Im(w * P)
                b_re[s] = br_[j][s];                 // Re(Q)
                b_im[s] = bi_[j][s];                 // -Im(Q)
                // advance P by exp(dtA * 256) for the next tile
                cmul(pwr[j][s], pwi[j][s], sr[j][s], si[j][s]);
            }
            // K-chunk j   : Re(wP) *  Re(Q)
            // K-chunk j+8 : Im(wP) * -Im(Q)   (sign folded into b_im)
            c = __builtin_amdgcn_wmma_f32_16x16x4_f32(false, a_re, false, b_re,
                                                      (short)0, c, false, false);
            c = __builtin_amdgcn_wmma_f32_16x16x4_f32(false, a_im, false, b_im,
                                                      (short)0, c, false, false);
        }

        // D layout: VGPR r holds (M = r + 8*half, N = idx) of the 16x16 tile.
        float* po = out + (size_t)h * LLEN + tau * 256 + half * 128 + idx;
#pragma unroll
        for (int r = 0; r < 8; ++r) {
            po[r * 16] = c[r];
        }
    }
}

extern "C" void kernel_launch(void* const* d_in, const int* in_sizes, int n_in,
                              void* d_out, int out_size, void* d_ws, size_t ws_size,
                              hipStream_t stream) {
    const float* C_ri       = (const float*)d_in[0];   // (H, 32, 2)
    const float* log_dt     = (const float*)d_in[1];   // (H,)
    const float* log_A_real = (const float*)d_in[2];   // (H, 32)
    const float* A_imag     = (const float*)d_in[3];   // (H, 32)
    // d_in[4] is L (scalar, == 4096), compiled in as LLEN.
    float* out = (float*)d_out;                        // (H, 4096)

    s4d_wmma_kernel<<<HDIM, 128, 0, stream>>>(C_ri, log_dt, log_A_real, A_imag, out);
}